// pointnet_2_23656679867649
// MI455X (gfx1250) — compile-verified
//
#include <hip/hip_runtime.h>

typedef float v2f __attribute__((ext_vector_type(2)));
typedef float v8f __attribute__((ext_vector_type(8)));

#define BN_INV_F 0.99999500003749981f  /* 1/sqrt(1+1e-5) */

// ---------------------------------------------------------------------------
// Build l0_xyz (B,N,3) from xyz input (B,9,N): channels 0..2 transposed.
// ---------------------------------------------------------------------------
__global__ void k_build_l0xyz(const float* __restrict__ xyz, float* __restrict__ l0,
                              int Bv, int Nv) {
  int i = blockIdx.x * blockDim.x + threadIdx.x;
  if (i >= Bv * Nv) return;
  int b = i / Nv, n = i % Nv;
  const float* p = xyz + (size_t)b * 9 * Nv + n;
  l0[(size_t)i * 3 + 0] = p[0 * Nv];
  l0[(size_t)i * 3 + 1] = p[1 * Nv];
  l0[(size_t)i * 3 + 2] = p[2 * Nv];
}

// ---------------------------------------------------------------------------
// Farthest point sampling. One workgroup (256 threads) per batch.
// dist[] lives in LDS; argmax via LDS tree reduction (first-max tie-break).
// ---------------------------------------------------------------------------
__global__ __launch_bounds__(256)
void k_fps(const float* __restrict__ pts, int S, int npoint, int* __restrict__ out) {
  __shared__ float dist[4096];
  __shared__ float rv[256];
  __shared__ int   ri[256];
  int b = blockIdx.x, tid = threadIdx.x;
  const float* p = pts + (size_t)b * S * 3;
  for (int i = tid; i < S; i += 256) dist[i] = 1e10f;
  __syncthreads();
  int far = 0;
  for (int j = 0; j < npoint; ++j) {
    if (tid == 0) out[b * npoint + j] = far;
    float fx = p[far * 3 + 0], fy = p[far * 3 + 1], fz = p[far * 3 + 2];
    float best = -1.0f; int bi = 0x7fffffff;
    for (int i = tid; i < S; i += 256) {
      float dx = p[i * 3 + 0] - fx, dy = p[i * 3 + 1] - fy, dz = p[i * 3 + 2] - fz;
      float d = dx * dx + dy * dy + dz * dz;
      float nd = fminf(dist[i], d);
      dist[i] = nd;
      if (nd > best) { best = nd; bi = i; }
    }
    rv[tid] = best; ri[tid] = bi;
    __syncthreads();
    for (int s = 128; s > 0; s >>= 1) {
      if (tid < s) {
        float v2 = rv[tid + s]; int i2 = ri[tid + s];
        if (v2 > rv[tid] || (v2 == rv[tid] && i2 < ri[tid])) { rv[tid] = v2; ri[tid] = i2; }
      }
      __syncthreads();
    }
    far = ri[0];
    __syncthreads();
  }
}

// gather sampled centroid coords: nx[b,j,:] = pts[b, fidx[b,j], :]
__global__ void k_gather_xyz(const float* __restrict__ pts, const int* __restrict__ fidx,
                             float* __restrict__ nx, int Ssrc, int Snew, int Bv) {
  int i = blockIdx.x * blockDim.x + threadIdx.x;
  if (i >= Bv * Snew) return;
  int b = i / Snew;
  int gi = fidx[i];
  const float* s = pts + ((size_t)b * Ssrc + gi) * 3;
  nx[(size_t)i * 3 + 0] = s[0];
  nx[(size_t)i * 3 + 1] = s[1];
  nx[(size_t)i * 3 + 2] = s[2];
}

// ---------------------------------------------------------------------------
// Ball query: first <=32 indices (ascending) with sqdist <= r^2, padded with
// the first hit (center itself always qualifies).
// ---------------------------------------------------------------------------
__global__ void k_ballquery(const float* __restrict__ src, const float* __restrict__ nx,
                            int* __restrict__ gidx, int Ssrc, int Snew, int Bv, float r2) {
  int i = blockIdx.x * blockDim.x + threadIdx.x;
  if (i >= Bv * Snew) return;
  int b = i / Snew;
  float cx = nx[(size_t)i * 3], cy = nx[(size_t)i * 3 + 1], cz = nx[(size_t)i * 3 + 2];
  const float* p = src + (size_t)b * Ssrc * 3;
  int* o = gidx + (size_t)i * 32;
  int cnt = 0, first = -1;
  for (int n = 0; n < Ssrc && cnt < 32; ++n) {
    float dx = p[n * 3] - cx, dy = p[n * 3 + 1] - cy, dz = p[n * 3 + 2] - cz;
    if (dx * dx + dy * dy + dz * dz <= r2) {
      if (first < 0) first = n;
      o[cnt++] = n;
    }
  }
  for (; cnt < 32; ++cnt) o[cnt] = first;
}

// ---------------------------------------------------------------------------
// Build grouped GEMM input: row r = (b*Snew + j)*32 + s ->
// [dx,dy,dz, feat(Cf), zero pad to K4]. feat_mode 0: feat is (B,Cf,Nld)
// channel-major (raw xyz input); mode 1: feat is (B*Ssrc, Cf) row-major.
// ---------------------------------------------------------------------------
__global__ void k_group(const float* __restrict__ src_xyz, const float* __restrict__ nx,
                        const int* __restrict__ gidx, const float* __restrict__ feat,
                        int feat_mode, int Cf, int K4, int Ssrc, int Snew, int Bv,
                        int Nld, float* __restrict__ A) {
  int r = blockIdx.x * blockDim.x + threadIdx.x;
  if (r >= Bv * Snew * 32) return;
  int i = r >> 5;
  int b = i / Snew;
  int gi = gidx[r];
  const float* sp = src_xyz + ((size_t)b * Ssrc + gi) * 3;
  float* a = A + (size_t)r * K4;
  a[0] = sp[0] - nx[(size_t)i * 3 + 0];
  a[1] = sp[1] - nx[(size_t)i * 3 + 1];
  a[2] = sp[2] - nx[(size_t)i * 3 + 2];
  if (feat_mode == 0) {
    const float* f = feat + (size_t)b * Cf * Nld + gi;
    for (int c = 0; c < Cf; ++c) a[3 + c] = f[(size_t)c * Nld];
  } else {
    const float* f = feat + ((size_t)b * Ssrc + gi) * Cf;
    for (int c = 0; c < Cf; ++c) a[3 + c] = f[c];
  }
  for (int c = 3 + Cf; c < K4; ++c) a[c] = 0.f;
}

// zero-pad weight rows: Wp (K4 x Nc) from W (K x Nc)
__global__ void k_padW(const float* __restrict__ W, float* __restrict__ Wp,
                       int K, int K4, int Nc) {
  int i = blockIdx.x * blockDim.x + threadIdx.x;
  if (i >= K4 * Nc) return;
  int k = i / Nc;
  Wp[i] = (k < K) ? W[i] : 0.f;
}

__global__ void k_zero(float* __restrict__ p, size_t n) {
  size_t i = (size_t)blockIdx.x * blockDim.x + threadIdx.x;
  if (i < n) p[i] = 0.f;
}

// ---------------------------------------------------------------------------
// fp32 WMMA GEMM: Out[M,Nc] = epilogue(A[M,K4] @ Bw[K4,Nc]).
// Requires M % 256 == 0, Nc % 32 == 0, K4 % 4 == 0 (true for every call).
// Block = 8 waves stacked in M (256 rows) x 32 cols; wave tile 32x32 via
// 2x2 v_wmma_f32_16x16x4_f32.
//
// B staging: the K4 x 32 B-slice lives in LDS in K-pair-interleaved layout:
//   group t = k/2, stride 96 floats (padded from 64 so the upper half-wave,
//   which reads group t+1, lands on banks 32..63 -> conflict-free b64 reads):
//   Bs[t*96 + col*2 + j] = Bw[(2t+j)*Nc + n0 + col]
// Each B operand (v2f, K-adjacent pair) is then ONE ds_load_b64, no shuffles.
//
// Epilogue: optional (x+bias)*(gamma*BN_INV)+beta, bias-only, ReLU, and
// fused 32-row max-pool via uint atomicMax (post-ReLU values >= 0,
// destination pre-zeroed).
// ---------------------------------------------------------------------------
#define BS_STRIDE 96

__global__ __launch_bounds__(256)
void k_gemm_wmma(const float* __restrict__ A, const float* __restrict__ Bw,
                 float* __restrict__ Out, int M, int K4, int Nc,
                 const float* __restrict__ bias, const float* __restrict__ gamma,
                 const float* __restrict__ beta, int relu, int pool) {
  extern __shared__ float Bs[];   // (K4/2) * BS_STRIDE floats
  int wave = threadIdx.x >> 5;
  int lane = threadIdx.x & 31;
  int half = lane >> 4, ln = lane & 15;
  int m0 = blockIdx.x * 256 + wave * 32;
  int n0 = blockIdx.y * 32;

  // cooperative stage of B slice into interleaved layout
  for (int i = threadIdx.x; i < K4 * 32; i += 256) {
    int t = i >> 6, col = (i >> 1) & 31, j = i & 1;
    Bs[t * BS_STRIDE + col * 2 + j] = Bw[(size_t)(2 * t + j) * Nc + n0 + col];
  }
  __syncthreads();

  v8f acc00 = {}, acc01 = {}, acc10 = {}, acc11 = {};

  // A fragment rows: lanes 0-15 hold K{kk,kk+1}, lanes 16-31 hold K{kk+2,kk+3}
  const float* a0p = A + (size_t)(m0 + ln) * K4 + 2 * half;
  const float* a1p = a0p + (size_t)16 * K4;
  // B: group t = kk/2 + half, columns ln (b0) and ln+16 (b1)
  const float* bsp = Bs + half * BS_STRIDE + ln * 2;

  for (int kk = 0; kk < K4; kk += 4) {
    __builtin_prefetch(a0p + 16, 0, 3);
    v2f a0 = *(const v2f*)a0p;
    v2f a1 = *(const v2f*)a1p;
    v2f b0 = *(const v2f*)(bsp);        // cols ln
    v2f b1 = *(const v2f*)(bsp + 32);   // cols ln+16
    a0p += 4; a1p += 4; bsp += 2 * BS_STRIDE;
    acc00 = __builtin_amdgcn_wmma_f32_16x16x4_f32(false, a0, false, b0, (short)0, acc00, false, false);
    acc01 = __builtin_amdgcn_wmma_f32_16x16x4_f32(false, a0, false, b1, (short)0, acc01, false, false);
    acc10 = __builtin_amdgcn_wmma_f32_16x16x4_f32(false, a1, false, b0, (short)0, acc10, false, false);
    acc11 = __builtin_amdgcn_wmma_f32_16x16x4_f32(false, a1, false, b1, (short)0, acc11, false, false);
  }

  int nB0 = n0 + ln, nB1 = n0 + 16 + ln;
  auto epi = [&](v8f acc, int mt, int nn) {
#pragma unroll
    for (int r = 0; r < 8; ++r) {
      int m = m0 + mt * 16 + half * 8 + r;   // C/D layout: VGPR r -> M=r / r+8
      float v = acc[r];
      if (gamma)      v = (v + bias[nn]) * (gamma[nn] * BN_INV_F) + beta[nn];
      else if (bias)  v = v + bias[nn];
      if (relu) v = fmaxf(v, 0.f);
      if (pool) {
        atomicMax((unsigned int*)(Out + (size_t)(m >> 5) * Nc + nn), __float_as_uint(v));
      } else {
        Out[(size_t)m * Nc + nn] = v;
      }
    }
  };
  epi(acc00, 0, nB0);
  epi(acc01, 0, nB1);
  epi(acc10, 1, nB0);
  epi(acc11, 1, nB1);
}

// ---------------------------------------------------------------------------
// Feature propagation input builder: 3-NN inverse-distance interp of p2,
// optionally concatenated after p1. Zero pad to K4.
// ---------------------------------------------------------------------------
__global__ void k_fp_build(const float* __restrict__ xyz1, const float* __restrict__ xyz2,
                           const float* __restrict__ p1, int C1,
                           const float* __restrict__ p2, int C2,
                           int S1, int S2, int Bv, int K4, float* __restrict__ A) {
  int i = blockIdx.x * blockDim.x + threadIdx.x;
  if (i >= Bv * S1) return;
  int b = i / S1;
  float cx = xyz1[(size_t)i * 3], cy = xyz1[(size_t)i * 3 + 1], cz = xyz1[(size_t)i * 3 + 2];
  const float* q = xyz2 + (size_t)b * S2 * 3;
  float d0 = 1e30f, d1 = 1e30f, d2 = 1e30f; int i0 = 0, i1 = 0, i2 = 0;
  for (int n = 0; n < S2; ++n) {
    float dx = q[n * 3] - cx, dy = q[n * 3 + 1] - cy, dz = q[n * 3 + 2] - cz;
    float d = dx * dx + dy * dy + dz * dz;
    if (d < d0)      { d2 = d1; i2 = i1; d1 = d0; i1 = i0; d0 = d; i0 = n; }
    else if (d < d1) { d2 = d1; i2 = i1; d1 = d;  i1 = n; }
    else if (d < d2) { d2 = d;  i2 = n; }
  }
  float w0 = 1.f / (d0 + 1e-8f), w1 = 1.f / (d1 + 1e-8f), w2 = 1.f / (d2 + 1e-8f);
  float ws = w0 + w1 + w2; w0 /= ws; w1 /= ws; w2 /= ws;
  float* a = A + (size_t)i * K4;
  int off = 0;
  if (p1) { const float* f = p1 + (size_t)i * C1; for (int c = 0; c < C1; ++c) a[off++] = f[c]; }
  const float* f0 = p2 + ((size_t)b * S2 + i0) * C2;
  const float* f1 = p2 + ((size_t)b * S2 + i1) * C2;
  const float* f2 = p2 + ((size_t)b * S2 + i2) * C2;
  for (int c = 0; c < C2; ++c) a[off++] = w0 * f0[c] + w1 * f1[c] + w2 * f2[c];
  for (; off < K4; ++off) a[off] = 0.f;
}

// global max over N per (b, c): out[b*C+c] = max_n x[(b*N+n)*C + c]
__global__ void k_globalmax(const float* __restrict__ x, float* __restrict__ out,
                            int Nv, int C, int Bv) {
  int i = blockIdx.x * blockDim.x + threadIdx.x;
  if (i >= Bv * C) return;
  int b = i / C, c = i % C;
  const float* p = x + (size_t)b * Nv * C + c;
  float m = -1e30f;
  for (int n = 0; n < Nv; ++n) m = fmaxf(m, p[(size_t)n * C]);
  out[i] = m;
}

// out[b,c,n] = l0p[b,n,c]
__global__ void k_transpose_out(const float* __restrict__ l0p, float* __restrict__ out,
                                int Nv, int C, int Bv) {
  size_t i = (size_t)blockIdx.x * blockDim.x + threadIdx.x;
  size_t tot = (size_t)Bv * C * Nv;
  if (i >= tot) return;
  int n = (int)(i % Nv); size_t t = i / Nv;
  int c = (int)(t % C);  int b = (int)(t / C);
  out[i] = l0p[((size_t)b * Nv + n) * C + c];
}

// ---------------------------------------------------------------------------
// Host side
// ---------------------------------------------------------------------------
static inline void gemm(hipStream_t s, const float* A, const float* Bw, float* Out,
                        int M, int K4, int Nc, const float* bias, const float* g,
                        const float* bt, int relu, int pool) {
  dim3 grid(M / 256, Nc / 32);
  size_t shmem = (size_t)(K4 / 2) * BS_STRIDE * sizeof(float);
  k_gemm_wmma<<<grid, 256, shmem, s>>>(A, Bw, Out, M, K4, Nc, bias, g, bt, relu, pool);
}
static inline void zero(hipStream_t s, float* p, size_t n) {
  k_zero<<<(unsigned)((n + 255) / 256), 256, 0, s>>>(p, n);
}

extern "C" void kernel_launch(void* const* d_in, const int* in_sizes, int n_in,
                              void* d_out, int out_size, void* d_ws, size_t ws_size,
                              hipStream_t stream) {
  (void)in_sizes; (void)n_in; (void)out_size; (void)ws_size;
  const int Bv = 16, Nv = 4096;
  const float* xyz = (const float*)d_in[0];

  // params flattened: per-MLP {W...,b...,g...,bt...}
  auto F = [&](int i) { return (const float*)d_in[i]; };
  // sa1: 1..12, sa2: 13..24, sa3: 25..36, fp3: 37..44, fp2: 45..52,
  // fp1: 53..64, conv1: 65..66
  const float *s1W0 = F(1), *s1W1 = F(2), *s1W2 = F(3), *s1b0 = F(4), *s1b1 = F(5), *s1b2 = F(6),
              *s1g0 = F(7), *s1g1 = F(8), *s1g2 = F(9), *s1t0 = F(10), *s1t1 = F(11), *s1t2 = F(12);
  const float *s2W0 = F(13), *s2W1 = F(14), *s2W2 = F(15), *s2b0 = F(16), *s2b1 = F(17), *s2b2 = F(18),
              *s2g0 = F(19), *s2g1 = F(20), *s2g2 = F(21), *s2t0 = F(22), *s2t1 = F(23), *s2t2 = F(24);
  const float *s3W0 = F(25), *s3W1 = F(26), *s3W2 = F(27), *s3b0 = F(28), *s3b1 = F(29), *s3b2 = F(30),
              *s3g0 = F(31), *s3g1 = F(32), *s3g2 = F(33), *s3t0 = F(34), *s3t1 = F(35), *s3t2 = F(36);
  const float *f3W0 = F(37), *f3W1 = F(38), *f3b0 = F(39), *f3b1 = F(40),
              *f3g0 = F(41), *f3g1 = F(42), *f3t0 = F(43), *f3t1 = F(44);
  const float *f2W0 = F(45), *f2W1 = F(46), *f2b0 = F(47), *f2b1 = F(48),
              *f2g0 = F(49), *f2g1 = F(50), *f2t0 = F(51), *f2t1 = F(52);
  const float *f1W0 = F(53), *f1W1 = F(54), *f1W2 = F(55), *f1b0 = F(56), *f1b1 = F(57), *f1b2 = F(58),
              *f1g0 = F(59), *f1g1 = F(60), *f1g2 = F(61), *f1t0 = F(62), *f1t1 = F(63), *f1t2 = F(64);
  const float *cW = F(65), *cb = F(66);

  // workspace arena (floats); peak ~260 MB
  float* W = (float*)d_ws;
  size_t o = 0;
  auto alloc = [&](size_t n) { float* p = W + o; o += n; return p; };
  float* l0x  = alloc((size_t)Bv * 4096 * 3);
  float* l1x  = alloc((size_t)Bv * 1024 * 3);
  float* l2x  = alloc((size_t)Bv * 256 * 3);
  float* l3x  = alloc((size_t)Bv * 64 * 3);
  float* l1p  = alloc((size_t)Bv * 1024 * 64);
  float* l2p  = alloc((size_t)Bv * 256 * 128);
  float* l3p  = alloc((size_t)Bv * 64 * 256);
  float* l2pn = alloc((size_t)Bv * 256 * 256);
  float* l1pn = alloc((size_t)Bv * 1024 * 128);
  float* l0p  = alloc((size_t)Bv * 4096 * 128);
  float* xbuf = alloc((size_t)Bv * 4096 * 128);
  float* A0   = alloc((size_t)131072 * 68 > (size_t)65536 * 136 ? (size_t)131072 * 68 : (size_t)65536 * 136);
  float* T1   = alloc((size_t)524288 * 32);
  float* T2   = alloc((size_t)524288 * 32);
  float* Wpad = alloc(32768);
  int* fi1 = (int*)alloc(Bv * 1024);
  int* fi2 = (int*)alloc(Bv * 256);
  int* fi3 = (int*)alloc(Bv * 64);
  int* gi1 = (int*)alloc((size_t)Bv * 1024 * 32);
  int* gi2 = (int*)alloc((size_t)Bv * 256 * 32);
  int* gi3 = (int*)alloc((size_t)Bv * 64 * 32);

  k_build_l0xyz<<<(Bv * Nv + 255) / 256, 256, 0, stream>>>(xyz, l0x, Bv, Nv);

  // ---- SA1: 4096 -> 1024, r=0.1, MLP [12,32,32,64] ----
  k_fps<<<Bv, 256, 0, stream>>>(l0x, 4096, 1024, fi1);
  k_gather_xyz<<<(Bv * 1024 + 255) / 256, 256, 0, stream>>>(l0x, fi1, l1x, 4096, 1024, Bv);
  k_ballquery<<<(Bv * 1024 + 255) / 256, 256, 0, stream>>>(l0x, l1x, gi1, 4096, 1024, Bv, 0.01f);
  {
    int M = Bv * 1024 * 32;
    k_group<<<(M + 255) / 256, 256, 0, stream>>>(l0x, l1x, gi1, xyz, 0, 9, 12, 4096, 1024, Bv, Nv, A0);
    zero(stream, l1p, (size_t)Bv * 1024 * 64);
    gemm(stream, A0, s1W0, T1, M, 12, 32, s1b0, s1g0, s1t0, 1, 0);
    gemm(stream, T1, s1W1, T2, M, 32, 32, s1b1, s1g1, s1t1, 1, 0);
    gemm(stream, T2, s1W2, l1p, M, 32, 64, s1b2, s1g2, s1t2, 1, 1);
  }

  // ---- SA2: 1024 -> 256, r=0.2, MLP [67,64,64,128] ----
  k_fps<<<Bv, 256, 0, stream>>>(l1x, 1024, 256, fi2);
  k_gather_xyz<<<(Bv * 256 + 255) / 256, 256, 0, stream>>>(l1x, fi2, l2x, 1024, 256, Bv);
  k_ballquery<<<(Bv * 256 + 255) / 256, 256, 0, stream>>>(l1x, l2x, gi2, 1024, 256, Bv, 0.04f);
  {
    int M = Bv * 256 * 32;
    k_group<<<(M + 255) / 256, 256, 0, stream>>>(l1x, l2x, gi2, l1p, 1, 64, 68, 1024, 256, Bv, 0, A0);
    k_padW<<<(68 * 64 + 255) / 256, 256, 0, stream>>>(s2W0, Wpad, 67, 68, 64);
    zero(stream, l2p, (size_t)Bv * 256 * 128);
    gemm(stream, A0, Wpad, T1, M, 68, 64, s2b0, s2g0, s2t0, 1, 0);
    gemm(stream, T1, s2W1, T2, M, 64, 64, s2b1, s2g1, s2t1, 1, 0);
    gemm(stream, T2, s2W2, l2p, M, 64, 128, s2b2, s2g2, s2t2, 1, 1);
  }

  // ---- SA3: 256 -> 64, r=0.4, MLP [131,128,128,256] ----
  k_fps<<<Bv, 256, 0, stream>>>(l2x, 256, 64, fi3);
  k_gather_xyz<<<(Bv * 64 + 255) / 256, 256, 0, stream>>>(l2x, fi3, l3x, 256, 64, Bv);
  k_ballquery<<<(Bv * 64 + 255) / 256, 256, 0, stream>>>(l2x, l3x, gi3, 256, 64, Bv, 0.16f);
  {
    int M = Bv * 64 * 32;
    k_group<<<(M + 255) / 256, 256, 0, stream>>>(l2x, l3x, gi3, l2p, 1, 128, 132, 256, 64, Bv, 0, A0);
    k_padW<<<(132 * 128 + 255) / 256, 256, 0, stream>>>(s3W0, Wpad, 131, 132, 128);
    zero(stream, l3p, (size_t)Bv * 64 * 256);
    gemm(stream, A0, Wpad, T1, M, 132, 128, s3b0, s3g0, s3t0, 1, 0);
    gemm(stream, T1, s3W1, T2, M, 128, 128, s3b1, s3g1, s3t1, 1, 0);
    gemm(stream, T2, s3W2, l3p, M, 128, 256, s3b2, s3g2, s3t2, 1, 1);
  }

  // ---- FP3: interp l3 -> l2, concat l2p(128)+interp(256)=384, MLP [384,256,256] ----
  {
    int M = Bv * 256;
    k_fp_build<<<(M + 255) / 256, 256, 0, stream>>>(l2x, l3x, l2p, 128, l3p, 256, 256, 64, Bv, 384, A0);
    gemm(stream, A0, f3W0, T1, M, 384, 256, f3b0, f3g0, f3t0, 1, 0);
    gemm(stream, T1, f3W1, l2pn, M, 256, 256, f3b1, f3g1, f3t1, 1, 0);
  }

  // ---- FP2: interp l2 -> l1, concat l1p(64)+interp(256)=320, MLP [320,256,128] ----
  {
    int M = Bv * 1024;
    k_fp_build<<<(M + 255) / 256, 256, 0, stream>>>(l1x, l2x, l1p, 64, l2pn, 256, 1024, 256, Bv, 320, A0);
    gemm(stream, A0, f2W0, T1, M, 320, 256, f2b0, f2g0, f2t0, 1, 0);
    gemm(stream, T1, f2W1, l1pn, M, 256, 128, f2b1, f2g1, f2t1, 1, 0);
  }

  // ---- FP1: interp l1 -> l0, interp(128) only, MLP [128,128,128,128] ----
  {
    int M = Bv * 4096;
    k_fp_build<<<(M + 255) / 256, 256, 0, stream>>>(l0x, l1x, nullptr, 0, l1pn, 128, 4096, 1024, Bv, 128, A0);
    gemm(stream, A0, f1W0, T1, M, 128, 128, f1b0, f1g0, f1t0, 1, 0);
    gemm(stream, T1, f1W1, T2, M, 128, 128, f1b1, f1g1, f1t1, 1, 0);
    gemm(stream, T2, f1W2, l0p, M, 128, 128, f1b2, f1g2, f1t2, 1, 0);
  }

  // ---- conv1 + global max + transposed point features ----
  {
    int M = Bv * 4096;
    gemm(stream, l0p, cW, xbuf, M, 128, 128, cb, nullptr, nullptr, 0, 0);
    float* out = (float*)d_out;
    k_globalmax<<<(Bv * 128 + 255) / 256, 256, 0, stream>>>(xbuf, out, 4096, 128, Bv);
    size_t tot = (size_t)Bv * 128 * 4096;
    k_transpose_out<<<(unsigned)((tot + 255) / 256), 256, 0, stream>>>(l0p, out + Bv * 128, 4096, 128, Bv);
  }
}